// TwoLayerGNN_37168646979926
// MI455X (gfx1250) — compile-verified
//
#include <hip/hip_runtime.h>

typedef float v2f __attribute__((ext_vector_type(2)));
typedef float v8f __attribute__((ext_vector_type(8)));
typedef unsigned int v4u __attribute__((ext_vector_type(4)));
typedef int v4i __attribute__((ext_vector_type(4)));
typedef int v8i __attribute__((ext_vector_type(8)));

#define DIM 128
#define WAVES_PER_BLOCK 8
#define ROWS_PER_BLOCK (WAVES_PER_BLOCK * 16)
#define A_STRIDE 132   // 128 + 4 pad floats per A-tile row (TDM pad reproduces this)
#define WP_PITCH 288   // floats per k-pair row of W (256 data + 32 pad; 288%64==32)

__device__ __forceinline__ void atomAddF(float* p, float v) {
  __hip_atomic_fetch_add(p, v, __ATOMIC_RELAXED, __HIP_MEMORY_SCOPE_AGENT);
}

// ---------------------------------------------------------------------------
// Utility fills / degree
// ---------------------------------------------------------------------------
__global__ void fill_f32(float* __restrict__ p, float v, size_t n) {
  size_t i = (size_t)blockIdx.x * blockDim.x + threadIdx.x;
  size_t stride = (size_t)gridDim.x * blockDim.x;
  for (; i < n; i += stride) p[i] = v;
}

__global__ void deg_kernel(const int* __restrict__ dst, float* __restrict__ deg, int E) {
  int e = blockIdx.x * blockDim.x + threadIdx.x;
  if (e < E) atomAddF(&deg[dst[e]], 1.0f);
}

__global__ void rsqrt_inplace(float* __restrict__ p, int n) {
  int i = blockIdx.x * blockDim.x + threadIdx.x;
  if (i < n) p[i] = rsqrtf(fmaxf(p[i], 1.0f));
}

// ---------------------------------------------------------------------------
// WMMA fp32 GEMM:  H[N,128] = A[N,128] @ W[128,128]
// A tiles staged via the Tensor Data Mover (tensor_load_to_lds + s_wait_tensorcnt),
// W staged pair-interleaved so every B-fragment is one ds_load_b64.
// ---------------------------------------------------------------------------
__global__ __launch_bounds__(256) void gemm128_wmma(const float* __restrict__ A,
                                                    const float* __restrict__ W,
                                                    float* __restrict__ Hout, int N) {
  __shared__ float sWp[(DIM / 2) * WP_PITCH];               // 64*288*4 = 73,728 B
  __shared__ float sA[WAVES_PER_BLOCK * 16 * A_STRIDE];     // 8*16*132*4 = 67,584 B

  const int tid  = threadIdx.x;
  const int wave = tid >> 5;
  const int lane = tid & 31;
  const int row0 = blockIdx.x * ROWS_PER_BLOCK + wave * 16;

  // ---- Stage W pair-interleaved: sWp[(k/2)*WP_PITCH + 2*n + (k&1)] = W[k][n].
  // B-frag (k0,half,col) is then the aligned b64 at pair (k0/2+half), col.
  for (int i = tid; i < DIM * DIM / 4; i += 256) {
    int idx = i * 4;
    int k = idx / DIM, n = idx % DIM;
    const float4 w4 = *(const float4*)(W + idx);
    float* dp = &sWp[(k >> 1) * WP_PITCH + 2 * n + (k & 1)];
    dp[0] = w4.x; dp[2] = w4.y; dp[4] = w4.z; dp[6] = w4.w;
  }

  // ---- Stage this wave's 16x128 A strip.
  float* myA = &sA[wave * 16 * A_STRIDE];
  if (row0 + 16 <= N) {
    // Full tile: Tensor Data Mover 2D descriptor (ISA cdna5 §8).
    // tile 128x16 f32, dense rows (stride 128), LDS pad 4 DWORDs per 128 DWORDs
    // -> rows land at A_STRIDE=132 floats.
    unsigned long long ga = (unsigned long long)(A + (size_t)row0 * DIM);
    unsigned ldsOff = (unsigned)(unsigned long long)myA;  // low 32 bits = LDS byte addr
    v4u g0;
    g0.x = 1u;                                  // count=1 (valid user descriptor)
    g0.y = ldsOff;                              // lds_addr (bytes)
    g0.z = (unsigned)(ga & 0xFFFFFFFFull);      // global_addr[31:0]
    g0.w = (unsigned)((ga >> 32) & 0x01FFFFFFull) | (2u << 30);  // addr[56:32] | type=2
    v8i g1;
    g1.s0 = (2 << 16)       // data_size = 4B
          | (1 << 20)       // pad_enable
          | (6 << 22)       // pad_interval: 128 DWORDs
          | (3 << 25);      // pad_amount:   4 DWORDs
    g1.s1 = (int)(128u << 16);   // tensor_dim0[15:0]=128 (bits 63:48 of group)
    g1.s2 = (int)(16u << 16);    // tensor_dim0 hi=0 | tensor_dim1[15:0]=16
    g1.s3 = (int)(128u << 16);   // tensor_dim1 hi=0 | tile_dim0=128
    g1.s4 = 16;                  // tile_dim1=16 rows, tile_dim2=0
    g1.s5 = 128;                 // tensor_dim0_stride = 128 elements
    g1.s6 = 0;
    g1.s7 = 0;
    v4i gz4 = {0, 0, 0, 0};           // groups 2/3 unused (2D tensor)
    v8i gz8 = {0, 0, 0, 0, 0, 0, 0, 0};
    __builtin_amdgcn_tensor_load_to_lds(g0, g1, gz4, gz4, gz8, /*cpol=*/0);
    __builtin_amdgcn_s_wait_tensorcnt(0);
  } else {
    // Tail tile: guarded manual copy with zero-fill.
    for (int i = lane; i < 16 * (DIM / 4); i += 32) {
      int r = i / (DIM / 4);
      int c4 = (i % (DIM / 4)) * 4;
      int grow = row0 + r;
      float4 a4 = make_float4(0.f, 0.f, 0.f, 0.f);
      if (grow < N) a4 = *(const float4*)(A + (size_t)grow * DIM + c4);
      float* dp = &myA[r * A_STRIDE + c4];
      dp[0] = a4.x; dp[1] = a4.y; dp[2] = a4.z; dp[3] = a4.w;
    }
  }
  __syncthreads();

  v8f acc[8] = {};
  const int half = lane >> 4;    // 0: lanes 0-15, 1: lanes 16-31
  const int lr   = lane & 15;

  for (int k0 = 0; k0 < DIM; k0 += 4) {
    // A frag (ISA 16x4 f32): lanes 0-15 -> K=k0,k0+1; lanes 16-31 -> K=k0+2,k0+3
    v2f afrag = *(const v2f*)&myA[lr * A_STRIDE + k0 + 2 * half];
#pragma unroll
    for (int nt = 0; nt < 8; ++nt) {
      const int col = nt * 16 + lr;
      // B frag: one aligned b64 = { W[k0+2*half][col], W[k0+2*half+1][col] }
      v2f bfrag = *(const v2f*)&sWp[((k0 >> 1) + half) * WP_PITCH + 2 * col];
      acc[nt] = __builtin_amdgcn_wmma_f32_16x16x4_f32(
          false, afrag, false, bfrag, (short)0, acc[nt], false, false);
    }
  }

  // C/D layout: VGPR v -> row v (lanes 0-15) / row v+8 (lanes 16-31); col = nt*16+lr
#pragma unroll
  for (int nt = 0; nt < 8; ++nt) {
#pragma unroll
    for (int v = 0; v < 8; ++v) {
      int grow = row0 + v + 8 * half;
      if (grow < N) Hout[(size_t)grow * DIM + nt * 16 + lr] = acc[nt][v];
    }
  }
}

// ---------------------------------------------------------------------------
// Edge scatter: agg[dst] += h[src] * dis[src]*dis[dst]   (one block per edge)
// ---------------------------------------------------------------------------
__global__ __launch_bounds__(128) void scatter_edges(const float* __restrict__ h,
                                                     const int* __restrict__ src,
                                                     const int* __restrict__ dst,
                                                     const float* __restrict__ dis,
                                                     float* __restrict__ agg, int E) {
  int e = blockIdx.x;
  if (e >= E) return;
  int s = src[e], d = dst[e];
  float nrm = dis[s] * dis[d];
  int t = threadIdx.x;
  atomAddF(&agg[(size_t)d * DIM + t], h[(size_t)s * DIM + t] * nrm);
}

// out = relu(agg + h*dis^2 (self loop) + b)
__global__ __launch_bounds__(128) void finalize_relu(const float* __restrict__ h,
                                                     const float* __restrict__ agg,
                                                     const float* __restrict__ dis,
                                                     const float* __restrict__ b,
                                                     float* __restrict__ out, int N) {
  int i = blockIdx.x;
  if (i >= N) return;
  int t = threadIdx.x;
  float self = dis[i] * dis[i];
  float v = agg[(size_t)i * DIM + t] + h[(size_t)i * DIM + t] * self + b[t];
  out[(size_t)i * DIM + t] = fmaxf(v, 0.0f);
}

// global mean pool accumulation
__global__ __launch_bounds__(128) void pool_kernel(const float* __restrict__ act,
                                                   const int* __restrict__ batch,
                                                   float* __restrict__ sums,
                                                   float* __restrict__ cnt, int N) {
  int i = blockIdx.x;
  if (i >= N) return;
  int g = batch[i];
  int t = threadIdx.x;
  atomAddF(&sums[(size_t)g * DIM + t], act[(size_t)i * DIM + t]);
  if (t == 0) atomAddF(&cnt[g], 1.0f);
}

// out[g] = (sums[g]/max(cnt,1)) . Wh + bh
__global__ __launch_bounds__(128) void head_kernel(const float* __restrict__ sums,
                                                   const float* __restrict__ cnt,
                                                   const float* __restrict__ Wh,
                                                   const float* __restrict__ bh,
                                                   float* __restrict__ out) {
  int g = threadIdx.x;  // 128 graphs
  float c = fmaxf(cnt[g], 1.0f);
  float acc = 0.0f;
  for (int j = 0; j < DIM; ++j) acc += (sums[(size_t)g * DIM + j] / c) * Wh[j];
  out[g] = acc + bh[0];
}

// ---------------------------------------------------------------------------
extern "C" void kernel_launch(void* const* d_in, const int* in_sizes, int n_in,
                              void* d_out, int out_size, void* d_ws, size_t ws_size,
                              hipStream_t stream) {
  const float* x   = (const float*)d_in[0];
  const float* W1  = (const float*)d_in[1];
  const float* b1  = (const float*)d_in[2];
  const float* W2  = (const float*)d_in[3];
  const float* b2  = (const float*)d_in[4];
  const float* Wh  = (const float*)d_in[5];
  const float* bh  = (const float*)d_in[6];
  const int*   ei  = (const int*)d_in[7];
  const int*   bat = (const int*)d_in[8];

  const int N = in_sizes[0] / DIM;
  const int E = in_sizes[7] / 2;
  const int* src = ei;
  const int* dst = ei + E;

  float* ws = (float*)d_ws;
  size_t o = 0;
  float* dis = ws + o; o += (size_t)N;           // deg -> dis in place
  o = (o + 255) & ~(size_t)255;
  float* bufH   = ws + o; o += (size_t)N * DIM;  // GEMM output (pre-aggregation)
  float* bufAgg = ws + o; o += (size_t)N * DIM;  // scatter accumulator
  float* bufAct = ws + o; o += (size_t)N * DIM;  // post bias+relu activations
  float* sums   = ws + o; o += (size_t)DIM * DIM;
  float* cnt    = ws + o; o += (size_t)DIM;      // contiguous after sums

  const int gemmGrid = (N + ROWS_PER_BLOCK - 1) / ROWS_PER_BLOCK;
  const size_t NH = (size_t)N * DIM;

  // ---- degree / normalization ----
  fill_f32<<<512, 256, 0, stream>>>(dis, 1.0f, (size_t)N);           // self-loop
  deg_kernel<<<(E + 255) / 256, 256, 0, stream>>>(dst, dis, E);
  rsqrt_inplace<<<(N + 255) / 256, 256, 0, stream>>>(dis, N);

  // ---- layer 1 ----
  gemm128_wmma<<<gemmGrid, 256, 0, stream>>>(x, W1, bufH, N);
  fill_f32<<<2048, 256, 0, stream>>>(bufAgg, 0.0f, NH);
  scatter_edges<<<E, 128, 0, stream>>>(bufH, src, dst, dis, bufAgg, E);
  finalize_relu<<<N, 128, 0, stream>>>(bufH, bufAgg, dis, b1, bufAct, N);

  // ---- layer 2 ----
  gemm128_wmma<<<gemmGrid, 256, 0, stream>>>(bufAct, W2, bufH, N);
  fill_f32<<<2048, 256, 0, stream>>>(bufAgg, 0.0f, NH);
  scatter_edges<<<E, 128, 0, stream>>>(bufH, src, dst, dis, bufAgg, E);
  finalize_relu<<<N, 128, 0, stream>>>(bufH, bufAgg, dis, b2, bufAct, N);

  // ---- global mean pool + head ----
  fill_f32<<<64, 256, 0, stream>>>(sums, 0.0f, (size_t)DIM * DIM + DIM);  // sums+cnt
  pool_kernel<<<N, 128, 0, stream>>>(bufAct, bat, sums, cnt, N);
  head_kernel<<<1, 128, 0, stream>>>(sums, cnt, Wh, bh, (float*)d_out);
}